// clDiceLoss_41790031790301
// MI455X (gfx1250) — compile-verified
//
#include <hip/hip_runtime.h>
#include <cstdint>

// clDice loss, fully fused:
//  Kernel 1 (x2): 10 soft-skeletonize iterations in LDS (halo-20 tile),
//                 async-DMA of the "other" tensor tile overlapped with compute,
//                 WMMA-based block reduction of (sum skel, sum skel*other).
//  Kernel 2     : single-wave finalize -> scalar loss.

#define IMG   512
#define TILE  64
#define HALO  20            // 10 iterations x radius 2
#define LTS   (TILE + 2*HALO)   // 104
#define NITER 10

typedef __attribute__((ext_vector_type(2))) float v2f;
typedef __attribute__((ext_vector_type(8))) float v8f;

__global__ __launch_bounds__(256)
void skel_partial_kernel(const float* __restrict__ src,
                         const float* __restrict__ other,
                         float* __restrict__ partials)
{
    __shared__ float xs[LTS * LTS];                 // evolving x
    __shared__ float ms[LTS * LTS];                 // erosion intermediate
    __shared__ __align__(16) float ot[TILE * TILE]; // "other" tensor tile
    __shared__ float wred[16];                      // per-wave partials

    const int tid = threadIdx.x;
    const int b   = blockIdx.z;
    const int tr0 = blockIdx.y * TILE;
    const int tc0 = blockIdx.x * TILE;
    const float* img  = src   + (size_t)b * (IMG * IMG);
    const float* oimg = other + (size_t)b * (IMG * IMG);
    const float  INF  = __builtin_inff();

    // ---- 1) Kick off async DMA of the fully-in-bounds 64x64 "other" tile ----
    // 4096 floats / 256 threads = 16 floats = 4 x b128 per lane.
    {
        const int row = tid >> 2;              // 0..63
        const int col = (tid & 3) << 4;        // 0,16,32,48
        const float*  gp = oimg + (size_t)(tr0 + row) * IMG + (tc0 + col);
        const uint32_t lp = (uint32_t)(uintptr_t)(&ot[row * TILE + col]);
        asm volatile(
            "global_load_async_to_lds_b128 %0, %1, off\n\t"
            "global_load_async_to_lds_b128 %0, %1, off offset:16\n\t"
            "global_load_async_to_lds_b128 %0, %1, off offset:32\n\t"
            "global_load_async_to_lds_b128 %0, %1, off offset:48"
            :: "v"(lp), "v"(gp) : "memory");
    }

    // ---- 2) Load 104x104 halo tile; image-OOB cells = +INF (min ignores) ----
    for (int i = tid; i < LTS * LTS; i += 256) {
        const int r  = i / LTS;
        const int c  = i - r * LTS;
        const int gr = tr0 - HALO + r;
        const int gc = tc0 - HALO + c;
        float v = INF;
        if (gr >= 0 && gr < IMG && gc >= 0 && gc < IMG)
            v = img[gr * IMG + gc];
        xs[i] = v;
    }
    __syncthreads();

    // ---- 3) 10 fused skeleton iterations with shrinking valid region ----
    const int tx = tid & 31;   // 32 cols
    const int ty = tid >> 5;   // 8 rows
    for (int t = 0; t < NITER; ++t) {
        // phase 1: erosion m = minpool3(x) on inset-(2t+1); image-OOB -> -INF
        const int lo1 = 2 * t + 1, hi1 = LTS - 2 * t - 1;
        for (int r = lo1 + ty; r < hi1; r += 8) {
            const int gr = tr0 - HALO + r;
            const bool rin = (gr >= 0) && (gr < IMG);
            for (int c = lo1 + tx; c < hi1; c += 32) {
                const int gc = tc0 - HALO + c;
                float m;
                if (rin && gc >= 0 && gc < IMG) {
                    const float* p = &xs[(r - 1) * LTS + (c - 1)];
                    float a0 = fminf(fminf(p[0],       p[1]),       p[2]);
                    float a1 = fminf(fminf(p[LTS],     p[LTS + 1]), p[LTS + 2]);
                    float a2 = fminf(fminf(p[2 * LTS], p[2*LTS+1]), p[2*LTS+2]);
                    m = fminf(fminf(a0, a1), a2);
                } else {
                    m = -INF;  // dilation padding semantics
                }
                ms[r * LTS + c] = m;
            }
        }
        __syncthreads();
        // phase 2: x = relu(x - relu(maxpool3(m) - m)) on inset-(2t+2)
        const int lo2 = 2 * t + 2, hi2 = LTS - 2 * t - 2;
        for (int r = lo2 + ty; r < hi2; r += 8) {
            const int gr = tr0 - HALO + r;
            const bool rin = (gr >= 0) && (gr < IMG);
            for (int c = lo2 + tx; c < hi2; c += 32) {
                const int gc = tc0 - HALO + c;
                if (rin && gc >= 0 && gc < IMG) {
                    const float* p = &ms[(r - 1) * LTS + (c - 1)];
                    const float mc = p[LTS + 1];
                    float d0 = fmaxf(fmaxf(p[0],       p[1]),       p[2]);
                    float d1 = fmaxf(fmaxf(p[LTS],     mc),         p[LTS + 2]);
                    float d2 = fmaxf(fmaxf(p[2 * LTS], p[2*LTS+1]), p[2*LTS+2]);
                    const float dil = fmaxf(fmaxf(d0, d1), d2);
                    const float contour = fmaxf(dil - mc, 0.0f);
                    const int idx = r * LTS + c;
                    xs[idx] = fmaxf(xs[idx] - contour, 0.0f);
                }
            }
        }
        __syncthreads();
    }

    // ---- 4) Wait for async tile, then per-thread partial sums ----
    asm volatile("s_wait_asynccnt 0" ::: "memory");
    __syncthreads();

    float pS = 0.0f, pI = 0.0f;
    for (int i = tid; i < TILE * TILE; i += 256) {
        const int r = i >> 6, c = i & 63;
        const float s = xs[(HALO + r) * LTS + (HALO + c)];
        pS += s;
        pI += s * ot[i];
    }

    // ---- 5) Wave reduction through the matrix pipe (v_wmma_f32_16x16x4) ----
    // A[m][k] holds pS for lanes m and m+16 in its K slots; B = 0.5 splat, so
    // D[m][n] = pS(m) + pS(m+16) independent of B's exact lane layout.
    v2f aS = { pS, pS };
    v2f aI = { pI, pI };
    v2f bh = { 0.5f, 0.5f };
    v8f z  = {};
    v8f dS = __builtin_amdgcn_wmma_f32_16x16x4_f32(false, aS, false, bh,
                                                   (short)0, z, false, false);
    v8f dI = __builtin_amdgcn_wmma_f32_16x16x4_f32(false, aI, false, bh,
                                                   (short)0, z, false, false);
    float sS = dS[0]+dS[1]+dS[2]+dS[3]+dS[4]+dS[5]+dS[6]+dS[7];
    float sI = dI[0]+dI[1]+dI[2]+dI[3]+dI[4]+dI[5]+dI[6]+dI[7];
    sS += __shfl_xor(sS, 16, 32);   // halves hold M=0..7 vs M=8..15 sums
    sI += __shfl_xor(sI, 16, 32);

    const int wid  = tid >> 5;
    const int lane = tid & 31;
    if (lane == 0) { wred[wid] = sS; wred[8 + wid] = sI; }
    __syncthreads();
    if (tid == 0) {
        float S = 0.0f, I = 0.0f;
        for (int w = 0; w < 8; ++w) { S += wred[w]; I += wred[8 + w]; }
        const int blk = (blockIdx.z * gridDim.y + blockIdx.y) * gridDim.x + blockIdx.x;
        partials[2 * blk + 0] = S;
        partials[2 * blk + 1] = I;
    }
}

// One wave: fixed-order per-batch sums -> scalar loss (deterministic).
__global__ __launch_bounds__(32)
void finalize_kernel(const float* __restrict__ p0,
                     const float* __restrict__ p1,
                     float* __restrict__ out)
{
    const int b = threadIdx.x;          // 0..31 = batch index
    float S1 = 0.f, I1 = 0.f, S2 = 0.f, I2 = 0.f;
    for (int t = 0; t < 64; ++t) {
        const int blk = b * 64 + t;
        S1 += p0[2 * blk + 0];
        I1 += p0[2 * blk + 1];
        S2 += p1[2 * blk + 0];
        I2 += p1[2 * blk + 1];
    }
    const float eps = 1e-6f;
    const float iflat = (I1 + eps) / (S1 + eps);  // norm_intersection(cl_pred, target)
    const float tflat = (I2 + eps) / (S2 + eps);  // norm_intersection(t_skel, pred)
    float n = iflat * tflat;
    float d = iflat + tflat;
    for (int off = 16; off > 0; off >>= 1) {
        n += __shfl_xor(n, off, 32);
        d += __shfl_xor(d, off, 32);
    }
    if (threadIdx.x == 0) out[0] = 1.0f - 2.0f * n / d;
}

extern "C" void kernel_launch(void* const* d_in, const int* in_sizes, int n_in,
                              void* d_out, int out_size, void* d_ws, size_t ws_size,
                              hipStream_t stream) {
    const float* pred   = (const float*)d_in[0];
    const float* target = (const float*)d_in[1];
    float* p0 = (float*)d_ws;        // 2048 blocks * 2 floats = 16 KB
    float* p1 = p0 + 2 * 2048;       // second 16 KB

    dim3 grid(IMG / TILE, IMG / TILE, 32);   // 8 x 8 x 32 = 2048 blocks
    skel_partial_kernel<<<grid, 256, 0, stream>>>(pred, target, p0);
    skel_partial_kernel<<<grid, 256, 0, stream>>>(target, pred, p1);
    finalize_kernel<<<1, 32, 0, stream>>>(p0, p1, (float*)d_out);
}